// VACE_87729001988668
// MI455X (gfx1250) — compile-verified
//
#include <hip/hip_runtime.h>
#include <hip/hip_bf16.h>

#define BDIM 4096
#define KDIM 32000
#define KV4  (KDIM / 4)          // 8000 float4 per row
#define TPB  256
#define NSTEPS ((KV4 + TPB - 1) / TPB)   // 32 (last step partial: 64 lanes)

// ---------------- CDNA5 async global->LDS helpers ----------------
#if defined(__has_builtin)
#if __has_builtin(__builtin_amdgcn_global_load_async_to_lds_b128)
#define HAVE_ASYNC_LDS 1
#endif
#endif

typedef int v4i_t __attribute__((vector_size(16)));
typedef __attribute__((address_space(1))) v4i_t* gm_v4i_p;   // global (__device__) ptr
typedef __attribute__((address_space(3))) v4i_t* lds_v4i_p;  // LDS ptr

__device__ __forceinline__ void async_copy16(const void* g, void* l) {
#if defined(HAVE_ASYNC_LDS)
  void* gnc = const_cast<void*>(g);
  __builtin_amdgcn_global_load_async_to_lds_b128((gm_v4i_p)gnc, (lds_v4i_p)l, 0, 0);
#else
  *(float4*)l = *(const float4*)g;   // synchronous fallback
#endif
}

template <int N>
__device__ __forceinline__ void wait_async() {
#if defined(HAVE_ASYNC_LDS)
#if __has_builtin(__builtin_amdgcn_s_wait_asynccnt)
  __builtin_amdgcn_s_wait_asynccnt(N);
#else
  if constexpr (N == 0) asm volatile("s_wait_asynccnt 0x0" ::: "memory");
  else                  asm volatile("s_wait_asynccnt 0x1" ::: "memory");
#endif
#endif
}

// ---------------- online logsumexp push ----------------
__device__ __forceinline__ void lse_push(float& m, float& s, float z) {
  if (z <= m) {
    s += __expf(z - m);
  } else {
    s = s * __expf(m - z) + 1.0f;   // first push: m=-inf -> s=1, m=z
    m = z;
  }
}

// ---------------- kernel D (placed first so disasm snippet shows it):
// per-row online logsumexp with async global->LDS double buffering ----------------
__global__ __launch_bounds__(TPB) void vace_lse_kernel(
    const float* __restrict__ logits, const float* __restrict__ inv_tau,
    const int* __restrict__ target, float* __restrict__ partial) {
  __shared__ float tile[2][TPB * 4];
  __shared__ float wm[TPB / 32], wsum[TPB / 32];

  const int row = blockIdx.x;
  const int tid = threadIdx.x;
  const float4* __restrict__ grow = (const float4*)(logits + (size_t)row * KDIM);
  const float4* __restrict__ gtau = (const float4*)inv_tau;

  // Prefetch step 0 (all lanes in range: tid < 8000). Uniform issue per wave.
  async_copy16(grow + tid, &tile[0][tid * 4]);

  float m = -__builtin_inff();
  float s = 0.0f;

  for (int st = 0; st < NSTEPS; ++st) {
    if (st + 1 < NSTEPS) {
      int nxt = (st + 1) * TPB + tid;
      int ci  = (nxt < KV4) ? nxt : (KV4 - 1);   // clamp tail -> uniform EXEC
      async_copy16(grow + ci, &tile[(st + 1) & 1][tid * 4]);
      wait_async<1>();                           // tile `st` landed (newest may fly)
    } else {
      wait_async<0>();
    }
    __syncthreads();                             // everyone's chunk of tile visible

    int idx = st * TPB + tid;                    // float4 index in row
    if (idx < KV4) {
      float4 v  = *(const float4*)&tile[st & 1][tid * 4];
      float4 it = gtau[idx];                     // L2-resident table
      lse_push(m, s, v.x * it.x);
      lse_push(m, s, v.y * it.y);
      lse_push(m, s, v.z * it.z);
      lse_push(m, s, v.w * it.w);
    }
    __syncthreads();                             // buffer safe to overwrite (st+2)
  }

  // wave32 reduction of (m, s)
  #pragma unroll
  for (int off = 16; off > 0; off >>= 1) {
    float m2 = __shfl_xor(m, off);
    float s2 = __shfl_xor(s, off);
    float mn = fmaxf(m, m2);
    s = s * __expf(m - mn) + s2 * __expf(m2 - mn);
    m = mn;
  }
  int wid = tid >> 5, lane = tid & 31;
  if (lane == 0) { wm[wid] = m; wsum[wid] = s; }
  __syncthreads();
  if (tid == 0) {
    float M = wm[0], S = wsum[0];
    #pragma unroll
    for (int w = 1; w < TPB / 32; ++w) {
      float m2 = wm[w], s2 = wsum[w];
      float mn = fmaxf(M, m2);
      S = S * __expf(M - mn) + s2 * __expf(m2 - mn);
      M = mn;
    }
    float lse = M + logf(S);
    int   t   = target[row];
    float own = logits[(size_t)row * KDIM + t];
    partial[row] = lse - own * inv_tau[t];
  }
}

// ---------------- kernel A: zero scratch ----------------
__global__ void vace_zero_kernel(int* p, int n) {
  int i = blockIdx.x * blockDim.x + threadIdx.x;
  if (i < n) p[i] = 0;
}

// ---------------- kernel B: per-class n, sum, sumsq via atomics ----------------
__global__ __launch_bounds__(TPB) void vace_stats_kernel(
    const float* __restrict__ logits, const int* __restrict__ target,
    int* __restrict__ n_i, float* __restrict__ s_f, float* __restrict__ ss_f) {
  int b = blockIdx.x * blockDim.x + threadIdx.x;
  if (b >= BDIM) return;
  int t = target[b];
  float own = logits[(size_t)b * KDIM + t];
  atomicAdd(&n_i[t], 1);
  atomicAdd(&s_f[t], own);
  atomicAdd(&ss_f[t], own * own);
}

// ---------------- kernel C: EMA var -> tau -> inv_tau table ----------------
__global__ __launch_bounds__(TPB) void vace_tau_kernel(
    const float* __restrict__ var, const int* __restrict__ count,
    const int* __restrict__ n_i, const float* __restrict__ s_f,
    const float* __restrict__ ss_f, float* __restrict__ inv_tau) {
  int k = blockIdx.x * blockDim.x + threadIdx.x;
  if (k >= KDIM) return;
  int   n   = n_i[k];
  int   cnt = count[k];
  float v   = var[k];
  float new_var;
  if (n > 0) {
    float nf = (float)n;
    float bm = s_f[k] / nf;
    float bv = ss_f[k] / nf - bm * bm;          // biased batch var
    float cv = (cnt > 0) ? (0.9f * v + 0.1f * bv) : bv;
    new_var  = fmaxf(cv, 1e-12f);
  } else {
    new_var = v;                                 // class absent: state unchanged
  }
  float tau = fminf(fmaxf(1.0f + new_var, 0.5f), 2.0f);
  if (cnt + n == 0) tau = 1.0f;                  // never-seen class -> init_tau
  inv_tau[k] = 1.0f / fmaxf(tau, 1e-12f);
}

// ---------------- kernel E: deterministic mean over B rows ----------------
__global__ __launch_bounds__(TPB) void vace_reduce_kernel(
    const float* __restrict__ partial, float* __restrict__ out) {
  __shared__ float sm[TPB];
  int tid = threadIdx.x;
  float acc = 0.0f;
  for (int i = tid; i < BDIM; i += TPB) acc += partial[i];
  sm[tid] = acc;
  __syncthreads();
  for (int st = TPB / 2; st > 0; st >>= 1) {
    if (tid < st) sm[tid] += sm[tid + st];
    __syncthreads();
  }
  if (tid == 0) out[0] = sm[0] * (1.0f / (float)BDIM);
}

extern "C" void kernel_launch(void* const* d_in, const int* in_sizes, int n_in,
                              void* d_out, int out_size, void* d_ws, size_t ws_size,
                              hipStream_t stream) {
  const float* logits = (const float*)d_in[0];
  // d_in[1] = mean (unused: output depends only on var -> tau)
  const float* var    = (const float*)d_in[2];
  const int*   target = (const int*)d_in[3];
  const int*   count  = (const int*)d_in[4];
  float* out = (float*)d_out;

  char*  ws      = (char*)d_ws;
  int*   n_i     = (int*)  (ws + 0 * (size_t)KDIM * 4);
  float* s_f     = (float*)(ws + 1 * (size_t)KDIM * 4);
  float* ss_f    = (float*)(ws + 2 * (size_t)KDIM * 4);
  float* inv_tau = (float*)(ws + 3 * (size_t)KDIM * 4);
  float* partial = (float*)(ws + 4 * (size_t)KDIM * 4);

  // A: re-zero n/s/ss every call (graph replays, no re-poison between runs)
  int zwords = 3 * KDIM;
  vace_zero_kernel<<<(zwords + TPB - 1) / TPB, TPB, 0, stream>>>(n_i, zwords);
  // B: segmented stats
  vace_stats_kernel<<<BDIM / TPB, TPB, 0, stream>>>(logits, target, n_i, s_f, ss_f);
  // C: temperature table
  vace_tau_kernel<<<(KDIM + TPB - 1) / TPB, TPB, 0, stream>>>(var, count, n_i, s_f, ss_f, inv_tau);
  // D: streaming online logsumexp, one workgroup per row
  vace_lse_kernel<<<BDIM, TPB, 0, stream>>>(logits, inv_tau, target, partial);
  // E: mean
  vace_reduce_kernel<<<1, TPB, 0, stream>>>(partial, out);
}